// CRF_87256555585986
// MI455X (gfx1250) — compile-verified
//
#include <hip/hip_runtime.h>
#include <hip/hip_bf16.h>
#include <cstddef>

// Problem constants (fixed by the reference setup_inputs).
constexpr int NB = 512;    // batch
constexpr int LS = 1024;   // sequence length
constexpr int CC = 128;    // classes
constexpr int ST = CC - 2; // start_tag = 126
constexpr int ET = CC - 1; // end_tag   = 127
constexpr float FINF = -1000.0f;

typedef float v2f __attribute__((ext_vector_type(2)));
typedef float v8f __attribute__((ext_vector_type(8)));

// ---------------------------------------------------------------------------
// Kernel 1: preprocess transition matrix.
//   rmax[i] = max_j T[i,j]
//   EB[i*C+j] = exp(T[i,j] - rmax[i])     (this is also B^T layout: B[k=j][col=i])
//   endT[i] = T[end_tag, i]
// ---------------------------------------------------------------------------
__global__ __launch_bounds__(128) void crf_prep(const float* __restrict__ T,
                                                float* __restrict__ rmax,
                                                float* __restrict__ endT,
                                                float* __restrict__ EB) {
  const int i = threadIdx.x;  // 0..127, one row each
  float mx = -3.4e38f;
  for (int j = 0; j < CC; ++j) mx = fmaxf(mx, T[i * CC + j]);
  rmax[i] = mx;
  for (int j = 0; j < CC; ++j) EB[i * CC + j] = __expf(T[i * CC + j] - mx);
  endT[i] = T[ET * CC + i];
}

// ---------------------------------------------------------------------------
// Kernel 2: sentence score (trans_score + emis_score), one wave32 per sample.
// ---------------------------------------------------------------------------
__global__ __launch_bounds__(256) void crf_sent(const float* __restrict__ logits,
                                                const float* __restrict__ T,
                                                const int* __restrict__ labels,
                                                const unsigned char* __restrict__ mask,
                                                float* __restrict__ sent) {
  const int wave = (int)((blockIdx.x * blockDim.x + threadIdx.x) >> 5);
  const int lane = threadIdx.x & 31;
  if (wave >= NB) return;
  const int n = wave;
  const int* lab = labels + (size_t)n * LS;
  const unsigned char* mrow = mask + (size_t)n * LS;
  const float* lrow = logits + (size_t)n * LS * CC;

  float s = 0.0f;
  // Pairs k = 0..L : trans_pair[k] = T[lab_ext[k+1], lab_ext[k]], weight mask_s[k]
  for (int k = lane; k <= LS; k += 32) {
    int prv = (k == 0) ? ST : (mrow[k - 1] ? lab[k - 1] : ET);
    int cur = (k == LS) ? ET : (mrow[k] ? lab[k] : ET);
    float wgt = (k == 0) ? 1.0f : (mrow[k - 1] ? 1.0f : 0.0f);
    s += wgt * T[cur * CC + prv];
    if (k < LS && mrow[k]) s += lrow[(size_t)k * CC + lab[k]];  // emission (raw label)
  }
  // wave32 reduce
  #pragma unroll
  for (int off = 16; off > 0; off >>= 1) s += __shfl_xor(s, off, 32);
  if (lane == 0) sent[n] = s;
}

// ---------------------------------------------------------------------------
// Kernel 3: forward algorithm via f32 WMMA + final NLL.
// One block = 16 batch rows, 8 waves; wave w owns output classes [16w,16w+16).
// alpha (16x128) lives in LDS; E tile (128x16) lives in 64 VGPRs per wave.
// ---------------------------------------------------------------------------
#define AS 132  // padded LDS row stride (floats): conflict-free ds_load_b64 pattern

__global__ __launch_bounds__(256) void crf_forward(const float* __restrict__ logits,
                                                   const unsigned char* __restrict__ mask,
                                                   const float* __restrict__ rmaxg,
                                                   const float* __restrict__ endTg,
                                                   const float* __restrict__ EB,
                                                   const float* __restrict__ sent,
                                                   float* __restrict__ out) {
  __shared__ float alpha[16 * AS];
  __shared__ float Pl[16 * AS];
  __shared__ float pm[256];
  __shared__ float psum[256];
  __shared__ float m_arr[16];
  __shared__ unsigned char mk[16];

  const int tid = threadIdx.x;
  const int w = tid >> 5;        // wave id 0..7 -> class tile
  const int lane = tid & 31;
  const int cc = lane & 15;      // N-column within tile / B column
  const int hi = lane >> 4;      // half-wave selector
  const int n0 = blockIdx.x * 16;
  const int iCol = 16 * w + cc;  // global class column for this lane

  // Preload B fragments for all 32 K-chunks (resident across all 1024 steps).
  // B chunk kc: VGPR0 = K=4kc+2hi, VGPR1 = K=4kc+2hi+1, column = iCol.
  v2f breg[32];
  #pragma unroll
  for (int kc = 0; kc < 32; ++kc)
    breg[kc] = *(const v2f*)(EB + (size_t)iCol * CC + 4 * kc + 2 * hi);
  const float rm = rmaxg[iCol];

  const int rown = tid >> 4;  // 0..15  (row for reduction phases)
  const int seg = tid & 15;   // 0..15  (8-wide column segment)

  // Init alpha
  #pragma unroll
  for (int q = 0; q < 8; ++q) {
    int j = 8 * seg + q;
    alpha[rown * AS + j] = (j == ST) ? 0.0f : FINF;
  }

  for (int t = 0; t < LS; ++t) {
    __syncthreads();  // alpha stable from previous epilogue
    // Phase A: partial row max (each thread reduces 8 columns)
    float mx = -3.4e38f;
    #pragma unroll
    for (int q = 0; q < 8; ++q) mx = fmaxf(mx, alpha[rown * AS + 8 * seg + q]);
    pm[tid] = mx;
    __syncthreads();
    // Phase B: finish max (redundant per row), compute P = exp(alpha - m)
    float m = pm[rown * 16];
    #pragma unroll
    for (int s = 1; s < 16; ++s) m = fmaxf(m, pm[rown * 16 + s]);
    if (seg == 0) m_arr[rown] = m;
    #pragma unroll
    for (int q = 0; q < 8; ++q) {
      int j = 8 * seg + q;
      Pl[rown * AS + j] = __expf(alpha[rown * AS + j] - m);
    }
    if (tid < 16) mk[tid] = mask[(size_t)(n0 + tid) * LS + t];
    __syncthreads();
    // Phase C: S = P @ E^T via 32x V_WMMA_F32_16X16X4_F32 (4 independent chains)
    v8f accs[4];
    #pragma unroll
    for (int c = 0; c < 4; ++c) accs[c] = (v8f){0.f, 0.f, 0.f, 0.f, 0.f, 0.f, 0.f, 0.f};
    #pragma unroll
    for (int kc = 0; kc < 32; ++kc) {
      v2f a = *(const v2f*)(Pl + cc * AS + 4 * kc + 2 * hi);
      accs[kc & 3] = __builtin_amdgcn_wmma_f32_16x16x4_f32(
          false, a, false, breg[kc], (short)0, accs[kc & 3], false, false);
    }
    v8f acc = (accs[0] + accs[1]) + (accs[2] + accs[3]);
    // Epilogue: alpha' = logit + log(S) + m + rmax, masked update.
    #pragma unroll
    for (int r = 0; r < 8; ++r) {
      const int M = r + 8 * hi;  // batch row within tile (C/D layout)
      float lg = logits[((size_t)(n0 + M) * LS + t) * CC + iCol];
      float nv = lg + __logf(acc[r]) + m_arr[M] + rm;
      float ov = alpha[M * AS + iCol];
      alpha[M * AS + iCol] = (mk[M] != 0) ? nv : ov;
    }
  }

  // Final: total = logsumexp_i(alpha + endT); out = total - sent
  __syncthreads();
  float fmx = -3.4e38f;
  #pragma unroll
  for (int q = 0; q < 8; ++q) {
    int j = 8 * seg + q;
    fmx = fmaxf(fmx, alpha[rown * AS + j] + endTg[j]);
  }
  pm[tid] = fmx;
  __syncthreads();
  float fm = pm[rown * 16];
  #pragma unroll
  for (int s = 1; s < 16; ++s) fm = fmaxf(fm, pm[rown * 16 + s]);
  float fs = 0.0f;
  #pragma unroll
  for (int q = 0; q < 8; ++q) {
    int j = 8 * seg + q;
    fs += __expf(alpha[rown * AS + j] + endTg[j] - fm);
  }
  psum[tid] = fs;
  __syncthreads();
  if (seg == 0) {
    float S = 0.0f;
    #pragma unroll
    for (int s = 0; s < 16; ++s) S += psum[rown * 16 + s];
    const int n = n0 + rown;
    out[n] = fm + __logf(S) - sent[n];
  }
}

// ---------------------------------------------------------------------------
extern "C" void kernel_launch(void* const* d_in, const int* in_sizes, int n_in,
                              void* d_out, int out_size, void* d_ws, size_t ws_size,
                              hipStream_t stream) {
  const float* logits = (const float*)d_in[0];
  const float* T = (const float*)d_in[1];
  const int* labels = (const int*)d_in[2];
  const unsigned char* mask = (const unsigned char*)d_in[3];  // jnp bool -> 1 byte
  float* out = (float*)d_out;

  // Workspace layout (floats): sent[512] | rmax[128] | endT[128] | EB[128*128]
  float* ws = (float*)d_ws;
  float* sent = ws;
  float* rmax = ws + 512;
  float* endT = ws + 640;
  float* EB = ws + 768;

  crf_prep<<<1, 128, 0, stream>>>(T, rmax, endT, EB);
  crf_sent<<<(NB * 32 + 255) / 256, 256, 0, stream>>>(logits, T, labels, mask, sent);
  crf_forward<<<NB / 16, 256, 0, stream>>>(logits, mask, rmax, endT, EB, sent, out);
}